// TripletLoss_10393820856647
// MI455X (gfx1250) — compile-verified
//
#include <hip/hip_runtime.h>

// Problem constants from the reference
#define B_ROWS 8192
#define D_DIM  4096
#define MARGIN 0.5f

typedef __attribute__((ext_vector_type(2))) float v2f;
typedef __attribute__((ext_vector_type(8))) float v8f;

// ---------------------------------------------------------------------------
// Kernel 1: one block (8 wave32) per anchor row. Streaming gather + reduce.
// Memory-bound: 3 x 16KB coalesced row reads per block, mostly L2 hits
// (128MB feature matrix fits in 192MB L2). Direct b128 global loads; no LDS
// staging since every element is used exactly once.
// ---------------------------------------------------------------------------
__global__ __launch_bounds__(256) void triplet_row_loss(
    const float* __restrict__ feats,
    const float* __restrict__ label,
    const int*   __restrict__ idx1,
    const int*   __restrict__ idx2,
    float*       __restrict__ row_loss)
{
    const int b    = blockIdx.x;
    const int t    = threadIdx.x;
    const int i1   = idx1[b];
    const int i2   = idx2[b];

    const float4* __restrict__ A  = (const float4*)(feats + (size_t)b  * D_DIM);
    const float4* __restrict__ T1 = (const float4*)(feats + (size_t)i1 * D_DIM);
    const float4* __restrict__ T2 = (const float4*)(feats + (size_t)i2 * D_DIM);

    float l11 = 0.0f, l12 = 0.0f, s1 = 0.0f, s2 = 0.0f;

    // 1024 float4 per row / 256 threads = 4 iterations, coalesced.
#pragma unroll
    for (int j = 0; j < 4; ++j) {
        const int k = t + j * 256;
        const float4 av  = A[k];
        const float4 t1v = T1[k];
        const float4 t2v = T2[k];

        float d;
        d = av.x - t1v.x; l11 += __builtin_fabsf(d); s1 = fmaf(d, d, s1);
        d = av.y - t1v.y; l11 += __builtin_fabsf(d); s1 = fmaf(d, d, s1);
        d = av.z - t1v.z; l11 += __builtin_fabsf(d); s1 = fmaf(d, d, s1);
        d = av.w - t1v.w; l11 += __builtin_fabsf(d); s1 = fmaf(d, d, s1);

        d = av.x - t2v.x; l12 += __builtin_fabsf(d); s2 = fmaf(d, d, s2);
        d = av.y - t2v.y; l12 += __builtin_fabsf(d); s2 = fmaf(d, d, s2);
        d = av.z - t2v.z; l12 += __builtin_fabsf(d); s2 = fmaf(d, d, s2);
        d = av.w - t2v.w; l12 += __builtin_fabsf(d); s2 = fmaf(d, d, s2);
    }

    // wave32 intra-wave reduction (warpSize == 32 on gfx1250)
#pragma unroll
    for (int off = 16; off > 0; off >>= 1) {
        l11 += __shfl_down(l11, off, 32);
        l12 += __shfl_down(l12, off, 32);
        s1  += __shfl_down(s1,  off, 32);
        s2  += __shfl_down(s2,  off, 32);
    }

    __shared__ float red[4][8];
    const int wave = t >> 5;
    const int lane = t & 31;
    if (lane == 0) {
        red[0][wave] = l11;
        red[1][wave] = l12;
        red[2][wave] = s1;
        red[3][wave] = s2;
    }
    __syncthreads();

    if (t == 0) {
        float L11 = 0.0f, L12 = 0.0f, S1 = 0.0f, S2 = 0.0f;
#pragma unroll
        for (int w = 0; w < 8; ++w) {
            L11 += red[0][w];
            L12 += red[1][w];
            S1  += red[2][w];
            S2  += red[3][w];
        }
        const bool swp = (L11 >= L12);           // t1 farther in L1 -> near = t2
        const float near_sse = swp ? S2 : S1;
        const float far_sse  = swp ? S1 : S2;

        const float lb0 = label[b];
        const float la  = label[i1];
        const float lb  = label[i2];
        const float near_l = swp ? lb : la;
        const float far_l  = swp ? la : lb;

        const float df = lb0 - far_l;
        const float dn = lb0 - near_l;
        const float alpha = df * df - dn * dn;

        float loss = near_sse - far_sse + alpha * MARGIN;
        row_loss[b] = loss > 0.0f ? loss : 0.0f;
    }
}

// ---------------------------------------------------------------------------
// Kernel 2: single-wave final reduction of the 8192 per-row losses using
// V_WMMA_F32_16X16X4_F32 as a cross-lane f32 reduction engine.
//   A (16x4 f32, 2 VGPRs/lane) = 64 loss values per chunk
//   B (4x16 f32, 2 VGPRs/lane) = all ones  => D[m][n] = rowsum(A[m]) + C[m][n]
// After 128 accumulating chunks every column of C is identical; lane0 holds
// M=0..7 of column 0 and lane16 holds M=8..15, so two partial sums finish it.
// EXEC is all ones for every WMMA (uniform control flow, 32 active lanes).
// ---------------------------------------------------------------------------
__global__ __launch_bounds__(32) void final_reduce_wmma(
    const float* __restrict__ row_loss,
    float*       __restrict__ out)
{
    const int lane = threadIdx.x;

    v2f ones;
    ones.x = 1.0f;
    ones.y = 1.0f;

    v8f c = {};

    for (int chunk = 0; chunk < B_ROWS; chunk += 64) {
        v2f a;
        a.x = row_loss[chunk + lane];
        a.y = row_loss[chunk + 32 + lane];
        // 8 args: (neg_a, A, neg_b, B, c_mod, C, reuse_a, reuse_b)
        c = __builtin_amdgcn_wmma_f32_16x16x4_f32(
                /*neg_a=*/false, a,
                /*neg_b=*/false, ones,
                /*c_mod=*/(short)0, c,
                /*reuse_a=*/false, /*reuse_b=*/false);
    }

    float s = c[0] + c[1] + c[2] + c[3] + c[4] + c[5] + c[6] + c[7];
    const float other = __shfl(s, 16, 32);   // lane16's partial (M=8..15)
    if (lane == 0) {
        out[0] = s + other;
    }
}

// ---------------------------------------------------------------------------
extern "C" void kernel_launch(void* const* d_in, const int* in_sizes, int n_in,
                              void* d_out, int out_size, void* d_ws, size_t ws_size,
                              hipStream_t stream)
{
    const float* feats = (const float*)d_in[0];   // [B, D] f32
    const float* label = (const float*)d_in[1];   // [B]    f32
    const int*   idx1  = (const int*)d_in[2];     // [B]    i32
    const int*   idx2  = (const int*)d_in[3];     // [B]    i32

    float* row_loss = (float*)d_ws;               // 8192 f32 scratch
    float* out      = (float*)d_out;              // scalar

    triplet_row_loss<<<B_ROWS, 256, 0, stream>>>(feats, label, idx1, idx2, row_loss);
    final_reduce_wmma<<<1, 32, 0, stream>>>(row_loss, out);
}